// MambaBlock_62706522522031
// MI455X (gfx1250) — compile-verified
//
#include <hip/hip_runtime.h>
#include <math.h>

// Problem dims (fixed from the reference)
#define BB 4
#define LL 2048
#define DD 256
#define EE 512
#define NN 16
#define RR 16
#define KK 4
#define ML (BB * LL)        // 8192 rows for all GEMMs
#define TWO_E (2 * EE)      // 1024

typedef float v2f __attribute__((ext_vector_type(2)));
typedef float v8f __attribute__((ext_vector_type(8)));

__device__ __forceinline__ float silu_f(float x) {
  return x / (1.0f + __expf(-x));
}
__device__ __forceinline__ float softplus_f(float x) {
  // numerically stable: max(x,0) + log1p(exp(-|x|))
  return fmaxf(x, 0.0f) + log1pf(__expf(-fabsf(x)));
}

#define WMMA_F32(A, Bv, C) \
  __builtin_amdgcn_wmma_f32_16x16x4_f32(false, (A), false, (Bv), (short)0, (C), false, false)

// ---------------------------------------------------------------------------
// Y[M,N] = X[M,K] (row stride lda) * W[N,K]^T (row stride ldb)
// One wave computes a 32(M) x 64(N) tile: 2 A fragments x 4 B fragments,
// 8 WMMAs per K-step. 2-deep software pipeline (fragments for k and k+4 in
// registers while loads for k+8 / k+12 are in flight) so s_wait_loadcnt
// waits cover ~2 K-steps of matrix work. WMMAs are interleaved a0/a1 so both
// A fragments are live simultaneously (prevents the allocator from folding
// them into one register and serializing the a1 load).
// Requires K % 8 == 0 and K >= 8.
// ep==1: Y = softplus(acc + bias[n])
// ---------------------------------------------------------------------------
__global__ __launch_bounds__(256) void wmma_gemm_2x4(
    const float* __restrict__ X, int lda,
    const float* __restrict__ W, int ldb,
    float* __restrict__ Y, int ldy,
    int M, int N, int K,
    const float* __restrict__ bias, int ep) {
  const int lane = threadIdx.x & 31;
  const int wid  = blockIdx.x * (blockDim.x >> 5) + (threadIdx.x >> 5);
  const int nstrips = N >> 6;
  const int mt = wid / nstrips;          // 32-row tile index
  const int ns = wid % nstrips;
  if (mt * 32 >= M) return;              // wave-uniform: EXEC all-ones at WMMA

  const int idx = lane & 15;             // m (A) / n (B)
  const int khi = lane >> 4;             // selects K pair {0,1} vs {2,3}

  const float* a0p = X + (size_t)(mt * 32 + idx) * lda + 2 * khi;
  const float* a1p = a0p + (size_t)16 * lda;
  const int n0 = ns * 64 + idx;
  const float* b0p = W + (size_t)(n0 +  0) * ldb + 2 * khi;
  const float* b1p = W + (size_t)(n0 + 16) * ldb + 2 * khi;
  const float* b2p = W + (size_t)(n0 + 32) * ldb + 2 * khi;
  const float* b3p = W + (size_t)(n0 + 48) * ldb + 2 * khi;

  v8f c00 = {}, c01 = {}, c02 = {}, c03 = {};
  v8f c10 = {}, c11 = {}, c12 = {}, c13 = {};

#define LD_FRAGS(va0, va1, vb0, vb1, vb2, vb3, koff)        \
  v2f va0 = *(const v2f*)(a0p + (koff));                    \
  v2f va1 = *(const v2f*)(a1p + (koff));                    \
  v2f vb0 = *(const v2f*)(b0p + (koff));                    \
  v2f vb1 = *(const v2f*)(b1p + (koff));                    \
  v2f vb2 = *(const v2f*)(b2p + (koff));                    \
  v2f vb3 = *(const v2f*)(b3p + (koff));

#define DO_STEP(va0, va1, vb0, vb1, vb2, vb3)               \
  c00 = WMMA_F32(va0, vb0, c00);                            \
  c10 = WMMA_F32(va1, vb0, c10);                            \
  c01 = WMMA_F32(va0, vb1, c01);                            \
  c11 = WMMA_F32(va1, vb1, c11);                            \
  c02 = WMMA_F32(va0, vb2, c02);                            \
  c12 = WMMA_F32(va1, vb2, c12);                            \
  c03 = WMMA_F32(va0, vb3, c03);                            \
  c13 = WMMA_F32(va1, vb3, c13);

  // prologue: fragments for k = 0 and k = 4 (2 steps in flight)
  LD_FRAGS(a0c, a1c, b0c, b1c, b2c, b3c, 0)
  LD_FRAGS(a0n, a1n, b0n, b1n, b2n, b3n, 4)

  for (int k = 8; k < K; k += 8) {
    {
      LD_FRAGS(a0f, a1f, b0f, b1f, b2f, b3f, k)
      DO_STEP(a0c, a1c, b0c, b1c, b2c, b3c)
      a0c = a0f; a1c = a1f;
      b0c = b0f; b1c = b1f; b2c = b2f; b3c = b3f;
    }
    {
      LD_FRAGS(a0g, a1g, b0g, b1g, b2g, b3g, k + 4)
      DO_STEP(a0n, a1n, b0n, b1n, b2n, b3n)
      a0n = a0g; a1n = a1g;
      b0n = b0g; b1n = b1g; b2n = b2g; b3n = b3g;
    }
  }
  DO_STEP(a0c, a1c, b0c, b1c, b2c, b3c)
  DO_STEP(a0n, a1n, b0n, b1n, b2n, b3n)

#undef LD_FRAGS
#undef DO_STEP

#pragma unroll
  for (int r = 0; r < 8; ++r) {
    const int m0 = mt * 32 + r + 8 * khi;
    const int m1 = m0 + 16;
    const int nb = ns * 64 + idx;
    float u0 = c00[r], u1 = c01[r], u2 = c02[r], u3 = c03[r];
    float w0 = c10[r], w1 = c11[r], w2 = c12[r], w3 = c13[r];
    if (ep == 1) {
      const float g0 = bias[nb + 0], g1 = bias[nb + 16];
      const float g2 = bias[nb + 32], g3 = bias[nb + 48];
      u0 = softplus_f(u0 + g0); u1 = softplus_f(u1 + g1);
      u2 = softplus_f(u2 + g2); u3 = softplus_f(u3 + g3);
      w0 = softplus_f(w0 + g0); w1 = softplus_f(w1 + g1);
      w2 = softplus_f(w2 + g2); w3 = softplus_f(w3 + g3);
    }
    float* y0 = Y + (size_t)m0 * ldy + nb;
    float* y1 = Y + (size_t)m1 * ldy + nb;
    y0[ 0] = u0; y0[16] = u1; y0[32] = u2; y0[48] = u3;
    y1[ 0] = w0; y1[16] = w1; y1[32] = w2; y1[48] = w3;
  }
}

// Same GEMM, 32(M) x 16(N) tile per wave (for the narrow N=48 sel GEMM),
// 2-deep pipelined the same way. Requires K % 8 == 0, K >= 8.
__global__ __launch_bounds__(256) void wmma_gemm_2x1(
    const float* __restrict__ X, int lda,
    const float* __restrict__ W, int ldb,
    float* __restrict__ Y, int ldy,
    int M, int N, int K) {
  const int lane = threadIdx.x & 31;
  const int wid  = blockIdx.x * (blockDim.x >> 5) + (threadIdx.x >> 5);
  const int nstrips = N >> 4;
  const int mt = wid / nstrips;
  const int ns = wid % nstrips;
  if (mt * 32 >= M) return;

  const int idx = lane & 15;
  const int khi = lane >> 4;

  const float* a0p = X + (size_t)(mt * 32 + idx) * lda + 2 * khi;
  const float* a1p = a0p + (size_t)16 * lda;
  const float* bp  = W + (size_t)(ns * 16 + idx) * ldb + 2 * khi;

  v8f c0 = {}, c1 = {};

  v2f a0c = *(const v2f*)(a0p),     a1c = *(const v2f*)(a1p);
  v2f bc  = *(const v2f*)(bp);
  v2f a0n = *(const v2f*)(a0p + 4), a1n = *(const v2f*)(a1p + 4);
  v2f bn  = *(const v2f*)(bp + 4);

  for (int k = 8; k < K; k += 8) {
    {
      v2f a0f = *(const v2f*)(a0p + k);
      v2f a1f = *(const v2f*)(a1p + k);
      v2f bf  = *(const v2f*)(bp + k);
      c0 = WMMA_F32(a0c, bc, c0);
      c1 = WMMA_F32(a1c, bc, c1);
      a0c = a0f; a1c = a1f; bc = bf;
    }
    {
      v2f a0g = *(const v2f*)(a0p + k + 4);
      v2f a1g = *(const v2f*)(a1p + k + 4);
      v2f bg  = *(const v2f*)(bp + k + 4);
      c0 = WMMA_F32(a0n, bn, c0);
      c1 = WMMA_F32(a1n, bn, c1);
      a0n = a0g; a1n = a1g; bn = bg;
    }
  }
  c0 = WMMA_F32(a0c, bc, c0);
  c1 = WMMA_F32(a1c, bc, c1);
  c0 = WMMA_F32(a0n, bn, c0);
  c1 = WMMA_F32(a1n, bn, c1);

#pragma unroll
  for (int r = 0; r < 8; ++r) {
    const int m0 = mt * 32 + r + 8 * khi;
    Y[(size_t)m0 * ldy + ns * 16 + idx]        = c0[r];
    Y[(size_t)(m0 + 16) * ldy + ns * 16 + idx] = c1[r];
  }
}

// ---------------------------------------------------------------------------
// Causal depthwise conv (K=4, pad (3,0)) + bias + SiLU over the x_in half of xz
// xz: (B*L, 1024), x_in = cols [0,512). Output xc: (B*L, 512)
// ---------------------------------------------------------------------------
__global__ __launch_bounds__(256) void conv_silu(
    const float* __restrict__ xz, const float* __restrict__ cw,
    const float* __restrict__ cb, float* __restrict__ xc) {
  const int t = blockIdx.x * blockDim.x + threadIdx.x;  // over B*L*E
  const int e  = t & (EE - 1);
  const int bl = t >> 9;               // EE = 512
  const int l  = bl & (LL - 1);
  const float* xin = xz + (size_t)bl * TWO_E + e;   // element (b,l,e)
  float s = cb[e];
#pragma unroll
  for (int j = 0; j < KK; ++j) {
    const int ll = l - (KK - 1) + j;
    if (ll >= 0) s = fmaf(cw[e * KK + j], xin[(long)(j - (KK - 1)) * TWO_E], s);
  }
  xc[t] = silu_f(s);
}

// ---------------------------------------------------------------------------
// Selective scan. One wave handles 2 channels e; within each 16-lane half,
// lane idx carries state h[n], n = idx. Per step:
//   h = exp(dt*A[e,n])*h + dt*B[n]*xc ;  y = sum_n h*C[n]
// reduction over n via 4 shfl_xor hops (stays inside each 16-lane half).
// Epilogue fused: ys = (y + D[e]*xc) * silu(z)
// Operands 16 iterations ahead are prefetched (global_prefetch_b8) so the loads
// sit in L0/L2 before the dependent exp/FMA chain needs them.
// ---------------------------------------------------------------------------
__global__ __launch_bounds__(256) void scan_kernel(
    const float* __restrict__ sel, const float* __restrict__ dt,
    const float* __restrict__ xc, const float* __restrict__ xz,
    const float* __restrict__ A, const float* __restrict__ Dp,
    float* __restrict__ ys) {
  const int lane = threadIdx.x & 31;
  const int wid  = blockIdx.x * (blockDim.x >> 5) + (threadIdx.x >> 5); // 0..1023
  const int b     = wid >> 8;          // EE/2 = 256 waves per batch
  const int epair = wid & 255;
  const int idx = lane & 15;           // n
  const int khi = lane >> 4;           // which of the two channels
  const int e = epair * 2 + khi;

  const float Aen  = A[e * NN + idx];
  const float Dp_e = Dp[e];

  float h = 0.0f;
  for (int l = 0; l < LL; ++l) {
    const size_t base = (size_t)b * LL + l;
    // prefetch operands ~16 steps ahead (stays inside this batch's rows)
    if (l + 16 < LL) {
      const size_t pb = base + 16;
      __builtin_prefetch(&dt[pb * EE + e], 0, 1);
      __builtin_prefetch(&xc[pb * EE + e], 0, 1);
      __builtin_prefetch(&sel[pb * 48 + RR + idx], 0, 1);
      __builtin_prefetch(&xz[pb * TWO_E + EE + e], 0, 1);
    }
    const float dt_v = dt[base * EE + e];
    const float xc_v = xc[base * EE + e];
    const float Bm   = sel[base * 48 + RR + idx];
    const float Cm   = sel[base * 48 + RR + NN + idx];

    const float dA = __expf(dt_v * Aen);
    h = fmaf(dA, h, dt_v * Bm * xc_v);

    float p = h * Cm;
    p += __shfl_xor(p, 1);
    p += __shfl_xor(p, 2);
    p += __shfl_xor(p, 4);
    p += __shfl_xor(p, 8);   // all 16 lanes of each half now hold the sum

    if (idx == 0) {
      const float zv = xz[base * TWO_E + EE + e];
      const float yv = fmaf(Dp_e, xc_v, p);
      ys[base * EE + e] = yv * silu_f(zv);
    }
  }
}

// ---------------------------------------------------------------------------
extern "C" void kernel_launch(void* const* d_in, const int* in_sizes, int n_in,
                              void* d_out, int out_size, void* d_ws, size_t ws_size,
                              hipStream_t stream) {
  const float* x   = (const float*)d_in[0];  // (B,L,D)
  const float* ipw = (const float*)d_in[1];  // (2E,D)
  const float* cw  = (const float*)d_in[2];  // (E,1,K)
  const float* cb  = (const float*)d_in[3];  // (E)
  const float* sw  = (const float*)d_in[4];  // (R+2N, E) = (48,512)
  const float* dtw = (const float*)d_in[5];  // (E,R)
  const float* dtb = (const float*)d_in[6];  // (E)
  const float* A   = (const float*)d_in[7];  // (E,N)
  const float* Dp  = (const float*)d_in[8];  // (E)
  const float* ow  = (const float*)d_in[9];  // (D,E)
  float* out = (float*)d_out;                // (B,L,D)

  char* ws = (char*)d_ws;
  const size_t off_xz  = 0;                                // 32 MB
  const size_t off_xc  = off_xz  + (size_t)ML * TWO_E * 4; // 16 MB
  const size_t off_sel = off_xc  + (size_t)ML * EE * 4;    // 1.5 MB
  const size_t off_dt  = off_sel + (size_t)ML * 48 * 4;    // 16 MB
  const size_t off_ys  = off_dt  + (size_t)ML * EE * 4;    // 16 MB
  float* xz  = (float*)(ws + off_xz);
  float* xc  = (float*)(ws + off_xc);
  float* sel = (float*)(ws + off_sel);
  float* dt  = (float*)(ws + off_dt);
  float* ys  = (float*)(ws + off_ys);

  // 1) xz = x @ in_proj_w^T : M=8192, N=1024, K=256
  {
    const int waves = (ML / 32) * (TWO_E / 64);   // 4096
    wmma_gemm_2x4<<<waves / 8, 256, 0, stream>>>(x, DD, ipw, DD, xz, TWO_E,
                                                 ML, TWO_E, DD, nullptr, 0);
  }
  // 2) xc = silu(depthwise_causal_conv(x_in) + conv_b)
  conv_silu<<<(BB * LL * EE) / 256, 256, 0, stream>>>(xz, cw, cb, xc);

  // 3) sel = xc @ sel_w^T : M=8192, N=48, K=512
  {
    const int waves = (ML / 32) * (48 / 16);      // 768
    wmma_gemm_2x1<<<waves / 8, 256, 0, stream>>>(xc, EE, sw, EE, sel, 48,
                                                 ML, 48, EE);
  }
  // 4) dt = softplus(sel[:, :16] @ dt_w^T + dt_b) : M=8192, N=512, K=16 (lda=48)
  {
    const int waves = (ML / 32) * (EE / 64);      // 2048
    wmma_gemm_2x4<<<waves / 8, 256, 0, stream>>>(sel, 48, dtw, RR, dt, EE,
                                                 ML, EE, RR, dtb, 1);
  }
  // 5) selective scan (sequential in L, parallel over B*E)
  {
    const int waves = BB * (EE / 2);              // 1024
    scan_kernel<<<(waves * 32) / 256, 256, 0, stream>>>(sel, dt, xc, xz, A, Dp, ys);
  }
  // 6) out = ys @ out_w^T : M=8192, N=256, K=512
  {
    const int waves = (ML / 32) * (DD / 64);      // 1024
    wmma_gemm_2x4<<<waves / 8, 256, 0, stream>>>(ys, EE, ow, EE, out, DD,
                                                 ML, DD, EE, nullptr, 0);
  }
}